// SR_Loss_84327387890351
// MI455X (gfx1250) — compile-verified
//
#include <hip/hip_runtime.h>
#include <math.h>

typedef __attribute__((ext_vector_type(2))) float v2f;
typedef __attribute__((ext_vector_type(8))) float v8f;
typedef int v4i __attribute__((vector_size(16)));

#define EPSF 1e-8f
#define SHARP 100.0f

constexpr int N_OBJ   = 8192;
constexpr int M_SR    = 4096;
constexpr int N_FACES = 2048;

static __device__ __forceinline__ v8f wmma4(v2f a, v2f b) {
    v8f c = {};
    return __builtin_amdgcn_wmma_f32_16x16x4_f32(false, a, false, b,
                                                 (short)0, c, false, false);
}

// ---- CDNA5 async global->LDS copy (b128), with safe fallback -------------
static __device__ __forceinline__ void copy_b128_to_lds(const float4* g, float4* l) {
#if defined(__has_builtin) && __has_builtin(__builtin_amdgcn_global_load_async_to_lds_b128)
    __builtin_amdgcn_global_load_async_to_lds_b128(
        (__attribute__((address_space(1))) v4i*)g,
        (__attribute__((address_space(3))) v4i*)l, 0, 0);
#else
    *l = *g;
#endif
}

static __device__ __forceinline__ void async_wait0() {
#if defined(__has_builtin) && __has_builtin(__builtin_amdgcn_s_wait_asynccnt)
    __builtin_amdgcn_s_wait_asynccnt(0);
#else
    asm volatile("s_wait_asynccnt 0x0" ::: "memory");
#endif
}

// ---------------------------------------------------------------------------
// Kernel 0: pack sr points as (x, y, z, |q|^2); the norm rides in the WMMA
// K=3 slot so no cross-lane broadcasts are needed in the NN loop.
// ---------------------------------------------------------------------------
__global__ __launch_bounds__(256) void sr_prep(const float* __restrict__ sr,
                                               float* __restrict__ srPack) {
    const int m = blockIdx.x * 256 + threadIdx.x;
    if (m >= M_SR) return;
    const float x = sr[m * 3 + 0], y = sr[m * 3 + 1], z = sr[m * 3 + 2];
    ((float4*)srPack)[m] = make_float4(x, y, z, x * x + y * y + z * z);
}

// ---------------------------------------------------------------------------
// Kernel 1: WMMA nearest-neighbor search.
// A row m = (q, |q|^2) ; B col n = (-2*o, 1)  =>  D[m,n] = |q|^2 - 2 q.o
// Each lane loads exactly its K-half as one ds_load_b64 (float2 indexing),
// so the A operand needs zero selects.  sr pack staged via async-to-LDS.
// ---------------------------------------------------------------------------
__global__ __launch_bounds__(256) void nn_kernel(const float* __restrict__ obj,
                                                 const float* __restrict__ srPack,
                                                 float* __restrict__ dPack,
                                                 float* __restrict__ cmap) {
    extern __shared__ float4 ldsSr[];            // M_SR float4 = 64 KB
    const int tid  = threadIdx.x;
    const int lane = tid & 31;
    const int h    = lane >> 4;                  // K-half
    const int l    = lane & 15;
    const int wave = (blockIdx.x * 256 + tid) >> 5;
    const int n    = wave * 16 + l;              // obj index

    const float4* gp = (const float4*)srPack;
    for (int i = tid; i < M_SR; i += 256) copy_b128_to_lds(&gp[i], &ldsSr[i]);
    async_wait0();
    __syncthreads();

    const float ox = obj[n * 3 + 0];
    const float oy = obj[n * 3 + 1];
    const float oz = obj[n * 3 + 2];
    v2f b;  b.x = h ? -2.0f * oz : -2.0f * ox;
            b.y = h ?  1.0f      : -2.0f * oy;   // (-2o, 1)
    const float objNorm = ox * ox + oy * oy + oz * oz;

    const float2* lds2 = (const float2*)ldsSr;   // [m*2 + h] = lane's K-half
    const int laneOff  = l * 2 + h;

    float minVal = 3.4e38f;
    int   minIdx = 0;

    #pragma unroll 4
    for (int base = 0; base < M_SR; base += 16) {
        const float2 aq = lds2[base * 2 + laneOff];
        v2f a;  a.x = aq.x;  a.y = aq.y;         // (q.xy) or (q.z, |q|^2)
        v8f acc = wmma4(a, b);                   // |q|^2 - 2 q.o
        #pragma unroll
        for (int v = 0; v < 8; ++v) {
            const int mi = base + v + 8 * h;
            if (acc[v] < minVal) { minVal = acc[v]; minIdx = mi; }
        }
    }

    const float oVal = __shfl_xor(minVal, 16, 32);
    const int   oIdx = __shfl_xor(minIdx, 16, 32);
    if (oVal < minVal) { minVal = oVal; minIdx = oIdx; }

    if (h == 0) {
        float d2 = objNorm + minVal;
        d2 = d2 > 0.0f ? d2 : 0.0f;
        const float nnd = sqrtf(d2 + EPSF);
        cmap[n] = 2.0f / (1.0f + __expf(SHARP * nnd));

        const float4 q = ldsSr[minIdx];
        ((float4*)dPack)[n] = make_float4(q.x - ox, q.y - oy, q.z - oz, 0.0f);
    }
}

// ---------------------------------------------------------------------------
// Kernel 2: per-face precompute: (fn,-fn.v0), (g_i,-g_i.v_i) with g_i = fn x e_i
// ---------------------------------------------------------------------------
__global__ __launch_bounds__(256) void face_prep(const float* __restrict__ verts,
                                                 const int*   __restrict__ faces,
                                                 const float* __restrict__ fnorm,
                                                 float* __restrict__ facePack) {
    const int f = blockIdx.x * 256 + threadIdx.x;
    if (f >= N_FACES) return;

    const int i0 = faces[f * 3 + 0], i1 = faces[f * 3 + 1], i2 = faces[f * 3 + 2];
    const float v0x = verts[i0*3], v0y = verts[i0*3+1], v0z = verts[i0*3+2];
    const float v1x = verts[i1*3], v1y = verts[i1*3+1], v1z = verts[i1*3+2];
    const float v2x = verts[i2*3], v2y = verts[i2*3+1], v2z = verts[i2*3+2];
    const float nx = fnorm[f*3], ny = fnorm[f*3+1], nz = fnorm[f*3+2];

    const float hh = nx * v0x + ny * v0y + nz * v0z;

    const float e0x = v1x - v0x, e0y = v1y - v0y, e0z = v1z - v0z;
    const float e1x = v2x - v1x, e1y = v2y - v1y, e1z = v2z - v1z;
    const float e2x = v0x - v2x, e2y = v0y - v2y, e2z = v0z - v2z;

    const float g0x = ny*e0z - nz*e0y, g0y = nz*e0x - nx*e0z, g0z = nx*e0y - ny*e0x;
    const float g1x = ny*e1z - nz*e1y, g1y = nz*e1x - nx*e1z, g1z = nx*e1y - ny*e1x;
    const float g2x = ny*e2z - nz*e2y, g2y = nz*e2x - nx*e2z, g2z = nx*e2y - ny*e2x;

    const float b0 = g0x*v0x + g0y*v0y + g0z*v0z;
    const float b1 = g1x*v1x + g1y*v1y + g1z*v1z;
    const float b2 = g2x*v2x + g2y*v2y + g2z*v2z;

    float4* fp = (float4*)facePack;
    fp[f * 4 + 0] = make_float4(nx,  ny,  nz,  -hh);
    fp[f * 4 + 1] = make_float4(g0x, g0y, g0z, -b0);
    fp[f * 4 + 2] = make_float4(g1x, g1y, g1z, -b1);
    fp[f * 4 + 3] = make_float4(g2x, g2y, g2z, -b2);
}

// ---------------------------------------------------------------------------
// Kernel 3: WMMA ray-triangle parity; face pack staged in LDS (128 KB) via
// async-to-LDS.  Per-lane float2 indexing -> each A operand is one
// ds_load_b64, no selects.  8 WMMAs per 16x16 (face,ray) tile.
// ---------------------------------------------------------------------------
__global__ __launch_bounds__(256) void hits_kernel(const float* __restrict__ obj,
                                                   const float* __restrict__ dPack,
                                                   const float* __restrict__ facePack,
                                                   float* __restrict__ contrib) {
    extern __shared__ float4 ldsFace[];          // N_FACES*4 float4 = 128 KB
    const int tid  = threadIdx.x;
    const int lane = tid & 31;
    const int h    = lane >> 4;
    const int l    = lane & 15;
    const int wave = (blockIdx.x * 256 + tid) >> 5;
    const int n    = wave * 16 + l;              // ray index

    const float4* gp = (const float4*)facePack;
    for (int i = tid; i < N_FACES * 4; i += 256) copy_b128_to_lds(&gp[i], &ldsFace[i]);
    async_wait0();
    __syncthreads();

    const float ox = obj[n * 3 + 0];
    const float oy = obj[n * 3 + 1];
    const float oz = obj[n * 3 + 2];
    const float4 d4 = ((const float4*)dPack)[n];

    v2f bo;  bo.x = h ? oz   : ox;    bo.y = h ? 1.0f : oy;   // (o, 1)
    v2f bd;  bd.x = h ? d4.z : d4.x;  bd.y = h ? 0.0f : d4.y; // (dir, 0)

    const float2* lds2 = (const float2*)ldsFace; // [(m*4+j)*2 + h]
    const int laneOff  = l * 8 + h;              // (l*4)*2 + h

    int cnt = 0;

    for (int base = 0; base < N_FACES; base += 16) {
        const int o8 = base * 8 + laneOff;
        const float2 p0 = lds2[o8 + 0];
        const float2 p1 = lds2[o8 + 2];
        const float2 p2 = lds2[o8 + 4];
        const float2 p3 = lds2[o8 + 6];

        v2f aFn; aFn.x = p0.x; aFn.y = p0.y;
        v2f aG0; aG0.x = p1.x; aG0.y = p1.y;
        v2f aG1; aG1.x = p2.x; aG1.y = p2.y;
        v2f aG2; aG2.x = p3.x; aG2.y = p3.y;

        v8f Dden = wmma4(aFn, bd);   // fn.dir
        v8f Dnum = wmma4(aFn, bo);   // fn.o - h == -num
        v8f G0d  = wmma4(aG0, bd);
        v8f E0   = wmma4(aG0, bo);   // g0.o - b0
        v8f G1d  = wmma4(aG1, bd);
        v8f E1   = wmma4(aG1, bo);
        v8f G2d  = wmma4(aG2, bd);
        v8f E2   = wmma4(aG2, bo);

        #pragma unroll
        for (int v = 0; v < 8; ++v) {
            const float den  = Dden[v];
            const float ad   = fabsf(den);
            const float safe = (ad < EPSF) ? EPSF : den;
            const float t    = (-Dnum[v]) / safe;
            const bool inside = (E0[v] + t * G0d[v] >= 0.0f)
                              & (E1[v] + t * G1d[v] >= 0.0f)
                              & (E2[v] + t * G2d[v] >= 0.0f);
            if (inside & (t >= 0.0f) & (ad >= EPSF)) cnt++;
        }
    }

    cnt += __shfl_xor(cnt, 16, 32);
    if (h == 0) {
        const float mag2 = d4.x * d4.x + d4.y * d4.y + d4.z * d4.z;
        contrib[n] = (cnt & 1) ? mag2 : 0.0f;
    }
}

// ---------------------------------------------------------------------------
// Kernel 4: deterministic reduction  pen_dist = sqrt(sum contrib)
// ---------------------------------------------------------------------------
__global__ __launch_bounds__(256) void reduce_kernel(const float* __restrict__ contrib,
                                                     float* __restrict__ out) {
    __shared__ float sdata[8];
    const int tid = threadIdx.x;
    float s = 0.0f;
    for (int i = tid; i < N_OBJ; i += 256) s += contrib[i];
    #pragma unroll
    for (int off = 16; off > 0; off >>= 1) s += __shfl_down(s, off, 32);
    if ((tid & 31) == 0) sdata[tid >> 5] = s;
    __syncthreads();
    if (tid == 0) {
        float t = 0.0f;
        #pragma unroll
        for (int w = 0; w < 8; ++w) t += sdata[w];
        out[0] = sqrtf(t);
    }
}

// ---------------------------------------------------------------------------
extern "C" void kernel_launch(void* const* d_in, const int* in_sizes, int n_in,
                              void* d_out, int out_size, void* d_ws, size_t ws_size,
                              hipStream_t stream) {
    const float* obj   = (const float*)d_in[0];   // [8192,3]
    const float* sr    = (const float*)d_in[1];   // [4096,3]
    const float* verts = (const float*)d_in[2];   // [2048,3]
    const int*   faces = (const int*)  d_in[3];   // [2048,3]
    const float* fnorm = (const float*)d_in[4];   // [2048,3]
    float* out = (float*)d_out;                   // [1 + 8192]

    char* ws = (char*)d_ws;
    float* dPack    = (float*)ws;                                   // 128 KB
    float* facePack = (float*)(ws + 131072);                        // 128 KB
    float* srPack   = (float*)(ws + 131072 + 131072);               //  64 KB
    float* contrib  = (float*)(ws + 131072 + 131072 + 65536);       //  32 KB

    sr_prep     <<<M_SR / 256,  256, 0,          stream>>>(sr, srPack);
    nn_kernel   <<<N_OBJ / 128, 256, 64 * 1024,  stream>>>(obj, srPack, dPack, out + 1);
    face_prep   <<<N_FACES / 256, 256, 0,        stream>>>(verts, faces, fnorm, facePack);
    hits_kernel <<<N_OBJ / 128, 256, 128 * 1024, stream>>>(obj, dPack, facePack, contrib);
    reduce_kernel<<<1, 256, 0,                   stream>>>(contrib, out);
}